// NNUtil_GPVAE_70489003262408
// MI455X (gfx1250) — compile-verified
//
#include <hip/hip_runtime.h>

typedef float v2f __attribute__((ext_vector_type(2)));
typedef float v8f __attribute__((ext_vector_type(8)));

#define N_TOT   8192
#define D_DIM   64
#define K_SEL   32
#define NQ      (N_TOT - K_SEL)        // 8160 query rows
#define QT      16                     // query rows per wave tile
#define NTILES  (NQ / QT)              // 510 (exact)
#define WPB     8                      // waves per block
#define HSTRIDE 33                     // heap row stride (pad vs 64 banks)
#define DSTRIDE 17                     // dist tile row stride (pad)

// ---------------- kernel 1: row norms into workspace ----------------
__global__ __launch_bounds__(256)
void row_norms_kernel(const float* __restrict__ x, float* __restrict__ sq) {
  int r = blockIdx.x * blockDim.x + threadIdx.x;
  if (r >= N_TOT) return;
  const float4* p = (const float4*)(x + r * D_DIM);
  float acc = 0.f;
#pragma unroll
  for (int i = 0; i < D_DIM / 4; ++i) {
    float4 v = p[i];
    acc += v.x * v.x + v.y * v.y + v.z * v.z + v.w * v.w;
  }
  sq[r] = acc;
}

// ---------------- kernel 2: WMMA distances + fused top-k ----------------
__global__ __launch_bounds__(WPB * 32)
void knn_wmma_kernel(const float* __restrict__ x, const float* __restrict__ sq,
                     float* __restrict__ outD, long long* __restrict__ outI) {
  __shared__ float sDist[WPB][QT * DSTRIDE];
  __shared__ float sHeapD[WPB][QT * HSTRIDE];
  __shared__ int   sHeapI[WPB][QT * HSTRIDE];

  const int lane = threadIdx.x & 31;
  const int wave = threadIdx.x >> 5;
  const int tile = blockIdx.x * WPB + wave;
  if (tile >= NTILES) return;            // uniform per wave -> EXEC stays all-1s

  const int qbase = K_SEL + tile * QT;   // first global query row in tile
  const int half  = lane >> 4;           // 0: lanes 0-15, 1: lanes 16-31
  const int l16   = lane & 15;

  float* distT = sDist[wave];
  float* heapD = sHeapD[wave];
  int*   heapI = sHeapI[wave];

  // init per-row top-k buffers
  for (int i = lane; i < QT * HSTRIDE; i += 32) {
    heapD[i] = __builtin_inff();
    heapI[i] = -1;
  }
  asm volatile("s_wait_dscnt 0" ::: "memory");

  float curMax = __builtin_inff();       // meaningful in lanes 0..15 only
  int   maxPos = 0;

  // A fragments: 16x64 query tile, ISA 16x4 f32 layout per K-step s:
  // lanes 0-15 hold (K=4s, 4s+1) of row M=l16; lanes 16-31 hold (4s+2, 4s+3)
  v2f qf[D_DIM / 4];
  {
    const float* qrow = x + (size_t)(qbase + l16) * D_DIM + 2 * half;
#pragma unroll
    for (int s = 0; s < D_DIM / 4; ++s)
      qf[s] = *(const v2f*)(qrow + 4 * s);
  }
  // query norms needed by this lane's C rows (M = v + 8*half)
  float sqq[8];
#pragma unroll
  for (int v = 0; v < 8; ++v) sqq[v] = sq[qbase + v + 8 * half];

  // software-pipelined B fragments (B = anchor tile^T; identical striping)
  const size_t laneByteOff = (size_t)l16 * D_DIM + 2 * half;
  v2f bf[D_DIM / 4];
  {
    const float* brow = x + laneByteOff;          // j0 = 0
#pragma unroll
    for (int s = 0; s < D_DIM / 4; ++s)
      bf[s] = *(const v2f*)(brow + 4 * s);
  }

  for (int j0 = 0; j0 <= qbase; j0 += 16) {
    // two accumulation chains (even/odd K-steps) -> shallower RAW depth
    v8f c0 = {}, c1 = {};
#pragma unroll
    for (int s = 0; s < D_DIM / 4; s += 2) {
      c0 = __builtin_amdgcn_wmma_f32_16x16x4_f32(
             false, qf[s],     false, bf[s],     (short)0, c0, false, false);
      c1 = __builtin_amdgcn_wmma_f32_16x16x4_f32(
             false, qf[s + 1], false, bf[s + 1], (short)0, c1, false, false);
    }

    const float sqa = sq[j0 + l16];               // candidate norm, column n=l16

    // prefetch next tile's B fragments; overlaps with epilogue + selection
    if (j0 + 16 <= qbase) {                       // wave-uniform scalar branch
      const float* brow = x + (size_t)(j0 + 16) * D_DIM + laneByteOff;
#pragma unroll
      for (int s = 0; s < D_DIM / 4; ++s)
        bf[s] = *(const v2f*)(brow + 4 * s);
    }

    // dist epilogue: C layout -> lane l, vgpr v holds (M=v+8*half, N=l16)
#pragma unroll
    for (int v = 0; v < 8; ++v) {
      const int m = v + 8 * half;
      const int r = qbase + m;
      const int j = j0 + l16;
      float dot = c0[v] + c1[v];
      float d = fmaxf(sqq[v] + sqa - 2.0f * dot, 0.0f);
      d = (j < r) ? d : __builtin_inff();
      distT[m * DSTRIDE + l16] = d;
    }
    asm volatile("s_wait_dscnt 0" ::: "memory");

    // lanes 0..15: threshold insertion into own row's top-32 buffer
    if (lane < QT) {
      const int m = lane;
      float* hD = heapD + m * HSTRIDE;
      int*   hI = heapI + m * HSTRIDE;
#pragma unroll 1
      for (int n = 0; n < 16; ++n) {
        float d = distT[m * DSTRIDE + n];
        if (d < curMax) {
          hD[maxPos] = d;
          hI[maxPos] = j0 + n;
          float mx = hD[0];
          int   mp = 0;
#pragma unroll 1
          for (int i = 1; i < K_SEL; ++i) {
            float hv = hD[i];
            if (hv > mx) { mx = hv; mp = i; }
          }
          curMax = mx;
          maxPos = mp;
        }
      }
    }
    asm volatile("s_wait_dscnt 0" ::: "memory");
  }

  // final: selection-sort 32 entries ascending, emit dists + int64 indices
  if (lane < QT) {
    const int m = lane;
    float* hD = heapD + m * HSTRIDE;
    int*   hI = heapI + m * HSTRIDE;
    const long long t = (long long)(qbase - K_SEL + m);
#pragma unroll 1
    for (int i = 0; i < K_SEL; ++i) {
      float best = hD[i];
      int   bp = i;
#pragma unroll 1
      for (int u = i + 1; u < K_SEL; ++u) {
        float hv = hD[u];
        if (hv < best) { best = hv; bp = u; }
      }
      float td = hD[i]; hD[i] = hD[bp]; hD[bp] = td;
      int   ti = hI[i]; hI[i] = hI[bp]; hI[bp] = ti;
      outD[t * K_SEL + i] = hD[i];
      outI[t * K_SEL + i] = (long long)hI[i];
    }
  }
}

// ---------------- launcher ----------------
extern "C" void kernel_launch(void* const* d_in, const int* in_sizes, int n_in,
                              void* d_out, int out_size, void* d_ws, size_t ws_size,
                              hipStream_t stream) {
  (void)in_sizes; (void)n_in; (void)out_size; (void)ws_size;
  const float* x  = (const float*)d_in[0];
  float*       sq = (float*)d_ws;                       // 8192 floats of scratch
  float*       outD = (float*)d_out;                    // [NQ, K] ascending dists
  long long*   outI = (long long*)((float*)d_out + (size_t)NQ * K_SEL); // [NQ, K] int64

  row_norms_kernel<<<(N_TOT + 255) / 256, 256, 0, stream>>>(x, sq);

  const int blocks = (NTILES + WPB - 1) / WPB;          // 64
  knn_wmma_kernel<<<blocks, WPB * 32, 0, stream>>>(x, sq, outD, outI);
}